// GraphModule_4303557231018
// MI455X (gfx1250) — compile-verified
//
#include <hip/hip_runtime.h>
#include <hip/hip_bf16.h>

typedef __attribute__((ext_vector_type(2))) float v2f;
typedef __attribute__((ext_vector_type(8))) float v8f;

#define H 128
#define LN_EPS 1e-5f

// ---------------------------------------------------------------- zero fill
__global__ void zero_k(float* __restrict__ p, long long n) {
    long long i = (long long)blockIdx.x * blockDim.x + threadIdx.x;
    long long stride = (long long)gridDim.x * blockDim.x;
    for (; i < n; i += stride) p[i] = 0.0f;
}

// ---------------------------------------------------------------- degrees
__global__ void deg_k(const int* __restrict__ src, const int* __restrict__ dst,
                      float* __restrict__ outd, float* __restrict__ ind, int E) {
    int e = blockIdx.x * blockDim.x + threadIdx.x;
    if (e < E) {
        unsafeAtomicAdd(&outd[src[e]], 1.0f);
        unsafeAtomicAdd(&ind[dst[e]], 1.0f);
    }
}

__global__ void rsqrt_deg_k(const float* __restrict__ outd, const float* __restrict__ ind,
                            float* __restrict__ rs_out, float* __restrict__ rs_in, int N) {
    int i = blockIdx.x * blockDim.x + threadIdx.x;
    if (i < N) {
        rs_out[i] = rsqrtf(fmaxf(outd[i], 1.0f));
        rs_in[i]  = rsqrtf(fmaxf(ind[i],  1.0f));
    }
}

// ------------------------------------------- AGG[dst] += X[src] * rs_out[src]
// one wave = one edge; lane = 4-float chunk. Edge indices are wave-uniform ->
// broadcast through readfirstlane so they live in SGPRs.
__global__ void scatter_k(const int* __restrict__ src, const int* __restrict__ dst,
                          const float* __restrict__ X, const float* __restrict__ rs,
                          float* __restrict__ AGG, int E) {
    long long t = (long long)blockIdx.x * blockDim.x + threadIdx.x;
    int e = (int)(t >> 5);
    int c = (int)(t & 31);
    if (e >= E) return;
    int s = __builtin_amdgcn_readfirstlane(src[e]);
    int d = __builtin_amdgcn_readfirstlane(dst[e]);
    float sc = __builtin_amdgcn_readfirstlane(__float_as_uint(rs[s])) * 0.0f;  // placeholder avoided below
    sc = rs[s];
    float4 v = ((const float4*)(X + (size_t)s * H))[c];
    float* ap = AGG + (size_t)d * H + c * 4;
    unsafeAtomicAdd(ap + 0, v.x * sc);
    unsafeAtomicAdd(ap + 1, v.y * sc);
    unsafeAtomicAdd(ap + 2, v.z * sc);
    unsafeAtomicAdd(ap + 3, v.w * sc);
}

// ---------------------------------------------------------------- WMMA GEMM
// Y[:, col_off : col_off+HO] = [resid +] act( diag(rs) * X[N,128] @ W[128,HO] + b )
// One wave computes a full 16 x HO output strip: HO/16 resident accumulators,
// A fragment loaded once per K-chunk, W offsets are compile-time immediates.
template<int HO>
__global__ void gemm_wmma_k(const float* __restrict__ X, const float* __restrict__ rs,
                            const float* __restrict__ W, const float* __restrict__ bias,
                            const float* __restrict__ resid, float* __restrict__ Y,
                            int N, int ldy, int col_off, int relu) {
    constexpr int NT = HO / 16;                 // column tiles held in registers
    const int lane = threadIdx.x;               // 0..31, one wave per block
    const int lo = lane & 15;                   // M row (A) / N col (B,C,D)
    const int hi = lane >> 4;                   // K-pair select (A,B) / row-half (C,D)
    const int m0 = blockIdx.x * 16;

    int row_a = m0 + lo;
    int row_ld = row_a < N ? row_a : N - 1;     // clamp loads, keep EXEC all-ones
    float scale = rs ? rs[row_ld] : 1.0f;
    const float* xr = X + row_ld * H;

    v8f acc[NT];
    #pragma unroll
    for (int t = 0; t < NT; ++t) acc[t] = (v8f){0.f,0.f,0.f,0.f,0.f,0.f,0.f,0.f};

    #pragma unroll 4
    for (int k0 = 0; k0 < H; k0 += 4) {
        const int ka = k0 + 2 * hi;             // this lane-half's K pair
        v2f a;
        a.x = xr[ka]     * scale;               // A: 16x4, M=lo, K=ka,ka+1
        a.y = xr[ka + 1] * scale;
        const float* wr = W + ka * HO;          // compile-time row stride
        #pragma unroll
        for (int t = 0; t < NT; ++t) {
            v2f b;
            b.x = wr[t * 16 + lo];              // B: 4x16, K=ka
            b.y = wr[HO + t * 16 + lo];         //          K=ka+1
            acc[t] = __builtin_amdgcn_wmma_f32_16x16x4_f32(
                         false, a, false, b, (short)0, acc[t], false, false);
        }
    }

    #pragma unroll
    for (int t = 0; t < NT; ++t) {
        const int col = t * 16 + lo;
        const float bv = bias[col];
        #pragma unroll
        for (int r = 0; r < 8; ++r) {
            int row = m0 + r + 8 * hi;          // C/D: VGPR r -> rows r, r+8
            if (row < N) {
                float v = acc[t][r] + bv;
                if (relu) v = fmaxf(v, 0.0f);
                int idx = row * ldy + col_off + col;
                if (resid) v += resid[idx];
                Y[idx] = v;
            }
        }
    }
}

// ------------------------------------- out = feat + LayerNorm(Xc) * g + b
// one wave per row; lane handles 4 consecutive features
__global__ void ln_resid_k(const float* __restrict__ Xc, const float* __restrict__ feat,
                           const float* __restrict__ g, const float* __restrict__ b,
                           float* __restrict__ out, int N) {
    int wid = threadIdx.x >> 5;
    int lane = threadIdx.x & 31;
    int row = blockIdx.x * (blockDim.x >> 5) + wid;
    if (row >= N) return;

    float4 v = ((const float4*)(Xc + (size_t)row * H))[lane];
    float s  = v.x + v.y + v.z + v.w;
    float sq = v.x * v.x + v.y * v.y + v.z * v.z + v.w * v.w;
    #pragma unroll
    for (int off = 16; off >= 1; off >>= 1) {
        s  += __shfl_xor(s,  off, 32);
        sq += __shfl_xor(sq, off, 32);
    }
    float mu  = s * (1.0f / H);
    float var = sq * (1.0f / H) - mu * mu;
    float inv = rsqrtf(var + LN_EPS);

    float4 gf = ((const float4*)g)[lane];
    float4 bf = ((const float4*)b)[lane];
    float4 ff = ((const float4*)(feat + (size_t)row * H))[lane];
    float4 o;
    o.x = ff.x + (v.x - mu) * inv * gf.x + bf.x;
    o.y = ff.y + (v.y - mu) * inv * gf.y + bf.y;
    o.z = ff.z + (v.z - mu) * inv * gf.z + bf.z;
    o.w = ff.w + (v.w - mu) * inv * gf.w + bf.w;
    ((float4*)(out + (size_t)row * H))[lane] = o;
}

// ----------------------------------------------------------------------------
extern "C" void kernel_launch(void* const* d_in, const int* in_sizes, int n_in,
                              void* d_out, int out_size, void* d_ws, size_t ws_size,
                              hipStream_t stream) {
    const float* feat = (const float*)d_in[0];
    const int*   src  = (const int*)d_in[1];
    const int*   dst  = (const int*)d_in[2];
    const float* W1a = (const float*)d_in[3];  const float* b1a = (const float*)d_in[4];
    const float* W2a = (const float*)d_in[5];  const float* b2a = (const float*)d_in[6];
    const float* W1b = (const float*)d_in[7];  const float* b1b = (const float*)d_in[8];
    const float* W2b = (const float*)d_in[9];  const float* b2b = (const float*)d_in[10];
    const float* Wf1 = (const float*)d_in[11]; const float* bf1 = (const float*)d_in[12];
    const float* Wf2 = (const float*)d_in[13]; const float* bf2 = (const float*)d_in[14];
    const float* lng = (const float*)d_in[15]; const float* lnb = (const float*)d_in[16];

    const int N = in_sizes[0] / H;      // 50000
    const int E = in_sizes[1];          // 800000
    const int Hh = H / 2;               // 64

    const size_t NH = (size_t)N * H;
    float* ws     = (float*)d_ws;
    float* AGG    = ws;                 // N*H   shared aggregation buffer
    float* HA     = ws + NH;            // N*H   ha1, later hcat
    float* HB     = ws + 2 * NH;        // N*H   hb1, later h
    float* outd   = ws + 3 * NH;        // N
    float* ind    = outd + N;           // N
    float* rs_out = ind + N;            // N
    float* rs_in  = rs_out + N;         // N

    const int ZB = 2048, ZT = 256;
    dim3 scat_g((unsigned)(((long long)E * 32 + 255) / 256));
    dim3 gM((N + 15) / 16);             // one wave per 16-row strip
    dim3 ln_g((N + 7) / 8);

    // degrees -> rsqrt factors
    zero_k<<<ZB, ZT, 0, stream>>>(outd, 2LL * N);
    deg_k<<<(E + 255) / 256, 256, 0, stream>>>(src, dst, outd, ind, E);
    rsqrt_deg_k<<<(N + 255) / 256, 256, 0, stream>>>(outd, ind, rs_out, rs_in, N);

    // gconv layer 1 (aggregation shared by both branches)
    zero_k<<<ZB, ZT, 0, stream>>>(AGG, (long long)NH);
    scatter_k<<<scat_g, 256, 0, stream>>>(src, dst, feat, rs_out, AGG, E);
    gemm_wmma_k<128><<<gM, 32, 0, stream>>>(AGG, rs_in, W1a, b1a, nullptr, HA, N, H, 0, 1);
    gemm_wmma_k<128><<<gM, 32, 0, stream>>>(AGG, rs_in, W1b, b1b, nullptr, HB, N, H, 0, 1);

    // branch a, gconv layer 2 -> hcat[:, 0:64]  (reuse HA as hcat)
    zero_k<<<ZB, ZT, 0, stream>>>(AGG, (long long)NH);
    scatter_k<<<scat_g, 256, 0, stream>>>(src, dst, HA, rs_out, AGG, E);
    gemm_wmma_k<64><<<gM, 32, 0, stream>>>(AGG, rs_in, W2a, b2a, nullptr, HA, N, H, 0, 0);

    // branch b, gconv layer 2 -> hcat[:, 64:128]
    zero_k<<<ZB, ZT, 0, stream>>>(AGG, (long long)NH);
    scatter_k<<<scat_g, 256, 0, stream>>>(src, dst, HB, rs_out, AGG, E);
    gemm_wmma_k<64><<<gM, 32, 0, stream>>>(AGG, rs_in, W2b, b2b, nullptr, HA, N, H, Hh, 0);

    // h = features + LayerNorm(hcat)   (HB := h)
    ln_resid_k<<<ln_g, 256, 0, stream>>>(HA, feat, lng, lnb, HB, N);

    // FFN: out = h + relu(h @ Wf1 + bf1) @ Wf2 + bf2
    gemm_wmma_k<128><<<gM, 32, 0, stream>>>(HB, nullptr, Wf1, bf1, nullptr, AGG, N, H, 0, 1);
    gemm_wmma_k<128><<<gM, 32, 0, stream>>>(AGG, nullptr, Wf2, bf2, HB, (float*)d_out, N, H, 0, 0);
}